// GIN_85031762526245
// MI455X (gfx1250) — compile-verified
//
#include <hip/hip_runtime.h>

typedef __attribute__((ext_vector_type(2))) float v2f;
typedef __attribute__((ext_vector_type(8))) float v8f;

#define HDIM 64
#define WSTR 72   // LDS stride for weight matrices (2-row offset -> disjoint bank group)
#define ASTR 17   // LDS stride for transposed A/T tiles [k][m] (conflict-free frag reads)
#define WAVES 2   // waves (16-row tiles) per workgroup

// ---------------------------------------------------------------------------
// WMMA helpers (V_WMMA_F32_16X16X4_F32, wave32)
// ---------------------------------------------------------------------------

__device__ __forceinline__ void stage_weights(const float* __restrict__ W, float* sW,
                                              int tid, int nthreads) {
  for (int i = tid; i < HDIM * HDIM; i += nthreads) {
    int k = i >> 6, n = i & 63;
    sW[k * WSTR + n] = W[i];
  }
}

// sA: transposed A tile [k][m], stride ASTR.  sW: weights [k][n], stride WSTR.
// Computes 16x(NT*16) += 16x64 @ 64x(NT*16) with 16 k-steps of K=4.
template <int NT>
__device__ __forceinline__ void gemm16(const float* sA, const float* sW, int lane, v8f* acc) {
  const int m  = lane & 15;
  const int h2 = (lane >> 4) << 1;  // 0 for lanes 0-15, 2 for lanes 16-31
#pragma unroll
  for (int kk = 0; kk < 16; ++kk) {
    const int k0 = kk * 4 + h2;
    v2f a;
    a.x = sA[(k0 + 0) * ASTR + m];
    a.y = sA[(k0 + 1) * ASTR + m];
#pragma unroll
    for (int nt = 0; nt < NT; ++nt) {
      v2f b;
      b.x = sW[(k0 + 0) * WSTR + nt * 16 + m];
      b.y = sW[(k0 + 1) * WSTR + nt * 16 + m];
      acc[nt] = __builtin_amdgcn_wmma_f32_16x16x4_f32(
          false, a, false, b, (short)0, acc[nt], false, false);
    }
  }
}

// ---------------------------------------------------------------------------
// Fused GIN MLP: out = relu((h+agg)@W1 + b1)@W2 + b2, one 16-row tile per wave
// ---------------------------------------------------------------------------
__global__ __launch_bounds__(WAVES * 32) void gin_mlp_kernel(
    const float* __restrict__ h, const float* __restrict__ agg,
    const float* __restrict__ W1, const float* __restrict__ b1,
    const float* __restrict__ W2, const float* __restrict__ b2,
    float* __restrict__ out, int ntiles) {
  __shared__ float sW1[HDIM * WSTR];
  __shared__ float sW2[HDIM * WSTR];
  __shared__ float sA[WAVES][HDIM * ASTR];
  __shared__ float sT[WAVES][HDIM * ASTR];

  const int tid  = threadIdx.x;
  const int lane = tid & 31;
  const int wave = tid >> 5;

  stage_weights(W1, sW1, tid, WAVES * 32);
  stage_weights(W2, sW2, tid, WAVES * 32);
  __syncthreads();

  const int tile = blockIdx.x * WAVES + wave;
  if (tile >= ntiles) return;  // wave-uniform; EXEC stays all-ones inside

  const size_t row0 = (size_t)tile * 16;
  float* sAw = sA[wave];
  float* sTw = sT[wave];

  // Load 16x64 tile of (h + agg), store transposed [k][m] into LDS
#pragma unroll
  for (int j = 0; j < 8; ++j) {
    int idx4 = lane + j * 32;       // float4 index 0..255
    int r    = idx4 >> 4;           // row 0..15
    int c    = (idx4 & 15) << 2;    // col 0..60
    const float4 hv = *(const float4*)(h + (row0 + r) * HDIM + c);
    const float4 av = *(const float4*)(agg + (row0 + r) * HDIM + c);
    sAw[(c + 0) * ASTR + r] = hv.x + av.x;
    sAw[(c + 1) * ASTR + r] = hv.y + av.y;
    sAw[(c + 2) * ASTR + r] = hv.z + av.z;
    sAw[(c + 3) * ASTR + r] = hv.w + av.w;
  }

  const int m     = lane & 15;
  const int halfM = (lane >> 4) << 3;
  const v8f zero  = {0.f, 0.f, 0.f, 0.f, 0.f, 0.f, 0.f, 0.f};

  v8f acc[4] = {zero, zero, zero, zero};
  gemm16<4>(sAw, sW1, lane, acc);

  // bias + ReLU, store intermediate transposed for second GEMM
#pragma unroll
  for (int nt = 0; nt < 4; ++nt) {
    const float bias = b1[nt * 16 + m];
#pragma unroll
    for (int v = 0; v < 8; ++v) {
      float val = acc[nt][v] + bias;
      val       = fmaxf(val, 0.f);
      sTw[(nt * 16 + m) * ASTR + (halfM + v)] = val;
    }
  }

  v8f acc2[4] = {zero, zero, zero, zero};
  gemm16<4>(sTw, sW2, lane, acc2);

#pragma unroll
  for (int nt = 0; nt < 4; ++nt) {
    const float bias = b2[nt * 16 + m];
#pragma unroll
    for (int v = 0; v < 8; ++v) {
      out[(row0 + halfM + v) * HDIM + nt * 16 + m] = acc2[nt][v] + bias;
    }
  }
}

// ---------------------------------------------------------------------------
// Classifier: out[G,10] = relu(pooled@Wc1 + bc1)@Wc2 + bc2
// ---------------------------------------------------------------------------
__global__ __launch_bounds__(WAVES * 32) void cls_kernel(
    const float* __restrict__ pooled,
    const float* __restrict__ Wc1, const float* __restrict__ bc1,
    const float* __restrict__ Wc2, const float* __restrict__ bc2,
    float* __restrict__ out, int ntiles, int C) {
  __shared__ float sW1[HDIM * WSTR];
  __shared__ float sW2[HDIM * WSTR];
  __shared__ float sA[WAVES][HDIM * ASTR];
  __shared__ float sT[WAVES][HDIM * ASTR];

  const int tid  = threadIdx.x;
  const int lane = tid & 31;
  const int wave = tid >> 5;

  stage_weights(Wc1, sW1, tid, WAVES * 32);
  for (int i = tid; i < HDIM * HDIM; i += WAVES * 32) {  // zero-pad 64x10 -> 64x16
    int k = i >> 6, n = i & 63;
    sW2[k * WSTR + n] = (n < C) ? Wc2[k * C + n] : 0.f;
  }
  __syncthreads();

  const int tile = blockIdx.x * WAVES + wave;
  if (tile >= ntiles) return;

  const size_t row0 = (size_t)tile * 16;
  float* sAw = sA[wave];
  float* sTw = sT[wave];

#pragma unroll
  for (int j = 0; j < 8; ++j) {
    int idx4 = lane + j * 32;
    int r    = idx4 >> 4;
    int c    = (idx4 & 15) << 2;
    const float4 pv = *(const float4*)(pooled + (row0 + r) * HDIM + c);
    sAw[(c + 0) * ASTR + r] = pv.x;
    sAw[(c + 1) * ASTR + r] = pv.y;
    sAw[(c + 2) * ASTR + r] = pv.z;
    sAw[(c + 3) * ASTR + r] = pv.w;
  }

  const int m     = lane & 15;
  const int halfM = (lane >> 4) << 3;
  const v8f zero  = {0.f, 0.f, 0.f, 0.f, 0.f, 0.f, 0.f, 0.f};

  v8f acc[4] = {zero, zero, zero, zero};
  gemm16<4>(sAw, sW1, lane, acc);

#pragma unroll
  for (int nt = 0; nt < 4; ++nt) {
    const float bias = bc1[nt * 16 + m];
#pragma unroll
    for (int v = 0; v < 8; ++v) {
      float val = acc[nt][v] + bias;
      val       = fmaxf(val, 0.f);
      sTw[(nt * 16 + m) * ASTR + (halfM + v)] = val;
    }
  }

  v8f acc2[1] = {zero};  // only columns 0..15 needed (C=10)
  gemm16<1>(sTw, sW2, lane, acc2);

  const float bias = (m < C) ? bc2[m] : 0.f;
#pragma unroll
  for (int v = 0; v < 8; ++v) {
    if (m < C) out[(row0 + halfM + v) * C + m] = acc2[0][v] + bias;
  }
}

// ---------------------------------------------------------------------------
// Edge scatter: agg[dst] += h[src].  One wave per edge, float2 per lane.
// ---------------------------------------------------------------------------
__global__ void scatter_kernel(const float* __restrict__ h, const int* __restrict__ src,
                               const int* __restrict__ dst, float* __restrict__ agg, int E) {
  long long t = (long long)blockIdx.x * blockDim.x + threadIdx.x;
  int e = (int)(t >> 5);
  if (e >= E) return;
  int lane = (int)(t & 31);
  int s = src[e], d = dst[e];
  const float2 v = *(const float2*)(h + (size_t)s * HDIM + lane * 2);
  float* p = agg + (size_t)d * HDIM + lane * 2;
  unsafeAtomicAdd(p + 0, v.x);
  unsafeAtomicAdd(p + 1, v.y);
}

// Pooling: pooled[batch[n]] += h[n]
__global__ void pool_kernel(const float* __restrict__ h, const int* __restrict__ batch,
                            float* __restrict__ pooled, int N) {
  long long t = (long long)blockIdx.x * blockDim.x + threadIdx.x;
  int n = (int)(t >> 5);
  if (n >= N) return;
  int lane = (int)(t & 31);
  int g = batch[n];
  const float2 v = *(const float2*)(h + (size_t)n * HDIM + lane * 2);
  float* p = pooled + (size_t)g * HDIM + lane * 2;
  unsafeAtomicAdd(p + 0, v.x);
  unsafeAtomicAdd(p + 1, v.y);
}

__global__ void zero_kernel(float* __restrict__ p, long long n) {
  long long i      = (long long)blockIdx.x * blockDim.x + threadIdx.x;
  long long stride = (long long)gridDim.x * blockDim.x;
  for (; i < n; i += stride) p[i] = 0.f;
}

// ---------------------------------------------------------------------------
extern "C" void kernel_launch(void* const* d_in, const int* in_sizes, int n_in,
                              void* d_out, int out_size, void* d_ws, size_t ws_size,
                              hipStream_t stream) {
  const float* x    = (const float*)d_in[0];
  const int* ei     = (const int*)d_in[1];
  const int* batch  = (const int*)d_in[2];
  const float* W1a  = (const float*)d_in[3];  const float* b1a = (const float*)d_in[4];
  const float* W2a  = (const float*)d_in[5];  const float* b2a = (const float*)d_in[6];
  const float* W1b  = (const float*)d_in[7];  const float* b1b = (const float*)d_in[8];
  const float* W2b  = (const float*)d_in[9];  const float* b2b = (const float*)d_in[10];
  const float* W1c  = (const float*)d_in[11]; const float* b1c = (const float*)d_in[12];
  const float* W2c  = (const float*)d_in[13]; const float* b2c = (const float*)d_in[14];
  const float* Wc1  = (const float*)d_in[15]; const float* bc1 = (const float*)d_in[16];
  const float* Wc2  = (const float*)d_in[17]; const float* bc2 = (const float*)d_in[18];

  const int N = in_sizes[0] / HDIM;   // 100000
  const int E = in_sizes[1] / 2;      // 1600000
  const int C = 10;
  const int G = out_size / C;         // 128
  const int* src = ei;
  const int* dst = ei + E;

  float* ws       = (float*)d_ws;
  const size_t N64 = (size_t)N * HDIM;
  float* hA     = ws;
  float* hB     = ws + N64;
  float* agg    = ws + 2 * N64;
  float* pooled = ws + 3 * N64;

  const int ntiles    = N / 16;                       // 6250 (exact)
  const int mlp_grid  = (ntiles + WAVES - 1) / WAVES;
  const long long sth = (long long)E * 32;
  const int sc_grid   = (int)((sth + 255) / 256);
  const long long pth = (long long)N * 32;
  const int pl_grid   = (int)((pth + 255) / 256);

  auto layer = [&](const float* hin, const float* W1, const float* b1,
                   const float* W2, const float* b2, float* hout) {
    hipLaunchKernelGGL(zero_kernel, dim3(2048), dim3(256), 0, stream, agg, (long long)N64);
    hipLaunchKernelGGL(scatter_kernel, dim3(sc_grid), dim3(256), 0, stream, hin, src, dst, agg, E);
    hipLaunchKernelGGL(gin_mlp_kernel, dim3(mlp_grid), dim3(WAVES * 32), 0, stream,
                       hin, agg, W1, b1, W2, b2, hout, ntiles);
  };

  layer(x,  W1a, b1a, W2a, b2a, hA);
  layer(hA, W1b, b1b, W2b, b2b, hB);
  layer(hB, W1c, b1c, W2c, b2c, hA);

  hipLaunchKernelGGL(zero_kernel, dim3(64), dim3(256), 0, stream, pooled, (long long)G * HDIM);
  hipLaunchKernelGGL(pool_kernel, dim3(pl_grid), dim3(256), 0, stream, hA, batch, pooled, N);

  const int cls_tiles = G / 16;  // 8
  hipLaunchKernelGGL(cls_kernel, dim3((cls_tiles + WAVES - 1) / WAVES), dim3(WAVES * 32), 0,
                     stream, pooled, Wc1, bc1, Wc2, bc2, (float*)d_out, cls_tiles, C);
}